// AdaptiveHierarchicalPool_41601053229344
// MI455X (gfx1250) — compile-verified
//
#include <hip/hip_runtime.h>
#include <hip/hip_bf16.h>
#include <stdint.h>

// AdaptiveHierarchicalPool for MI455X (gfx1250).
// Bandwidth-bound: one full pass over x (512 MiB) is the cost floor (~23 us @
// 23.3 TB/s). We map the segment-sum einsum 'bst,btd->bsd' per batch onto
// V_WMMA_F32_16X16X4_F32: A = 16x4 one-hot segment indicator (from an LDS byte
// code table), B = 4x16 f32 tile of x streamed once from HBM, C = 16x16 f32
// accumulator covering segments 0..9 (padded to 16) x 16 D-columns.

#define NUM_SEG 10

typedef __attribute__((ext_vector_type(2))) float v2f;
typedef __attribute__((ext_vector_type(8))) float v8f;

__global__ __launch_bounds__(256) void seg_pool_wmma_kernel(
    const float* __restrict__ x,      // [B, T, D]
    const uint8_t* __restrict__ mask, // [B, T] (jax bool -> 1 byte)
    const int* __restrict__ length,   // [B]
    float* __restrict__ out,          // [B*S*D] seg_feat ++ [B*S] seg_mask
    int B, int T, int D)
{
    extern __shared__ uint8_t smem[];
    uint8_t* s_codes = smem;                              // T bytes
    float* s_cnt = (float*)(smem + ((T + 15) & ~15));     // 16 floats

    const int tid  = threadIdx.x;
    const int lane = tid & 31;
    const int wave = tid >> 5;                 // 0..7
    const int blocksPerBatch = D / 128;        // 8 waves * 16 cols
    const int b  = blockIdx.x / blocksPerBatch;
    const int tg = blockIdx.x % blocksPerBatch;

    if (tid < 16) s_cnt[tid] = 0.0f;
    __syncthreads();

    // ---- Phase 1: per-t segment codes + masked counts into LDS ----
    // bnd[s] = trunc(linspace(0, L, 11)[s]) = trunc(L * s / 10) (f32 math).
    const float Lf = (float)length[b];
    const uint8_t* mb = mask + (size_t)b * T;
    for (int t = tid; t < T; t += blockDim.x) {
        uint8_t code = 0xFF;
        if (mb[t]) {
            #pragma unroll
            for (int s = 0; s < NUM_SEG; ++s) {
                int lo = (int)(Lf * ((float)s * 0.1f));
                int hi = (int)(Lf * ((float)(s + 1) * 0.1f));
                if (t >= lo && t < hi) { code = (uint8_t)s; break; }
            }
        }
        s_codes[t] = code;                       // ds_store_b8
        if (code != 0xFF) atomicAdd(&s_cnt[code], 1.0f);  // ds_add_f32
    }
    __syncthreads();

    // ---- Phase 2: WMMA K-loop over T (each x element read exactly once) ----
    // A 16x4 f32 layout (ISA 7.12.2): lanes 0-15 hold M=lane, VGPR0=K0,
    // VGPR1=K1; lanes 16-31 hold M=lane-16, VGPR0=K2, VGPR1=K3.
    // B 4x16 f32 layout mirrors: VGPR0 lanes0-15 = row K0 (N=lane),
    // lanes16-31 = row K2; VGPR1 = rows K1/K3.
    const int col  = lane & 15;                 // M for A, N for B
    const int koff = (lane >> 4) << 1;          // 0 or 2
    const int n0   = tg * 128 + wave * 16;      // this wave's D-tile base
    const float* xb = x + (size_t)b * T * D + (n0 + col);

    v8f acc = {};
    for (int t = 0; t < T; t += 4) {
        const int r0 = t + koff;
        const uint8_t c0 = s_codes[r0];
        const uint8_t c1 = s_codes[r0 + 1];
        v2f a;
        a.x = (c0 == (uint8_t)col) ? 1.0f : 0.0f;
        a.y = (c1 == (uint8_t)col) ? 1.0f : 0.0f;

        const float* p = xb + (size_t)r0 * D;
        v2f bm;
        bm.x = p[0];                            // x[b, r0,   n0+col]
        bm.y = p[D];                            // x[b, r0+1, n0+col]
        __builtin_prefetch(p + (size_t)128 * D, 0, 3);  // global_prefetch_b8 lookahead

        // (neg_a, A, neg_b, B, c_mod, C, reuse_a, reuse_b)
        acc = __builtin_amdgcn_wmma_f32_16x16x4_f32(
            false, a, false, bm, (short)0, acc, false, false);
    }

    // ---- Phase 3: normalize by counts and store ----
    // C layout: VGPR r -> M=r (lanes 0-15) or M=r+8 (lanes 16-31), N=lane&15.
    const int mbase = (lane >> 4) << 3;         // 0 or 8
    #pragma unroll
    for (int r = 0; r < 8; ++r) {
        const int s = mbase + r;
        if (s < NUM_SEG) {
            const float cnt = s_cnt[s];
            const float v = (cnt > 0.0f) ? acc[r] / cnt : 0.0f;
            out[((size_t)b * NUM_SEG + s) * D + n0 + col] = v;
        }
    }
    // seg_mask (one wave per batch writes it), appended after seg_feat.
    if (tg == 0 && wave == 0 && lane < NUM_SEG) {
        out[(size_t)B * NUM_SEG * D + (size_t)b * NUM_SEG + lane] =
            (s_cnt[lane] > 0.0f) ? 1.0f : 0.0f;
    }
}

extern "C" void kernel_launch(void* const* d_in, const int* in_sizes, int n_in,
                              void* d_out, int out_size, void* d_ws, size_t ws_size,
                              hipStream_t stream) {
    (void)n_in; (void)out_size; (void)d_ws; (void)ws_size;
    const float*   x      = (const float*)d_in[0];
    const uint8_t* mask   = (const uint8_t*)d_in[1];
    const int*     length = (const int*)d_in[2];
    float*         out    = (float*)d_out;

    const int B  = in_sizes[2];
    const int BT = in_sizes[1];
    const int T  = BT / B;                 // 4096
    const int D  = in_sizes[0] / BT;       // 1024

    const int blocksPerBatch = D / 128;    // 8
    dim3 grid(B * blocksPerBatch);         // 256 blocks = 2048 waves
    dim3 block(256);                       // 8 wave32 waves
    size_t smem = (size_t)((T + 15) & ~15) + 16 * sizeof(float);

    hipLaunchKernelGGL(seg_pool_wmma_kernel, grid, block, smem, stream,
                       x, mask, length, out, B, T, D);
}